// Up_755914244166
// MI455X (gfx1250) — compile-verified
//
#include <hip/hip_runtime.h>
#include <cstdint>

// ---------------- problem constants ----------------
#define B_    8
#define INCH  128
#define OUTCH 64
#define NVv   40962
#define NVIN  10242
#define NFf   81920
#define EPSf  1e-5f
#define TPB   2561                 // ceil(NVv/16) tiles per batch
#define NTILES (B_*TPB)            // 20488
#define NWG_K2 1024
#define NWG_MC 1024

typedef float v2f __attribute__((ext_vector_type(2)));
typedef float v8f __attribute__((ext_vector_type(8)));

// D(16x16,f32) = A(16x4,f32) * B(4x16,f32) + C   — native CDNA5 f32 WMMA
static __device__ inline v8f wmma_f32_k4(v2f a, v2f b, v8f c) {
  return __builtin_amdgcn_wmma_f32_16x16x4_f32(false, a, false, b, (short)0, c, false, false);
}

// ---------------- K0: fold grad_w x {ew,ns} into per-face-corner weights ----------------
__global__ void k_prep_face(const float* __restrict__ grad_w, const float* __restrict__ ew,
                            const float* __restrict__ ns, float* __restrict__ gwe,
                            float* __restrict__ gwn) {
  int i = blockIdx.x * blockDim.x + threadIdx.x;
  if (i >= NFf * 3) return;
  int f = i / 3;
  float g0 = grad_w[i * 2 + 0], g1 = grad_w[i * 2 + 1];
  gwe[i] = g0 * ew[f * 2 + 0] + g1 * ew[f * 2 + 1];
  gwn[i] = g0 * ns[f * 2 + 0] + g1 * ns[f * 2 + 1];
}

// ---------------- K1: transpose x1 [B,64,NVIN] -> x1t [B,NVIN,64] ----------------
__global__ void __launch_bounds__(256) k_transpose_x1(const float* __restrict__ x1,
                                                      float* __restrict__ x1t) {
  __shared__ float tile[64][65];
  const int nb = (NVIN + 63) / 64;
  int b = blockIdx.x / nb;
  int v0 = (blockIdx.x % nb) * 64;
  int t = threadIdx.x;
  for (int i = 0; i < 16; i++) {
    int c = i * 4 + (t >> 6), v = t & 63;
    if (v0 + v < NVIN) tile[v][c] = x1[((size_t)b * 64 + c) * NVIN + v0 + v];
  }
  __syncthreads();
  for (int i = 0; i < 16; i++) {
    int v = i * 4 + (t >> 6), c = t & 63;
    if (v0 + v < NVIN) x1t[((size_t)b * NVIN + v0 + v) * 64 + c] = tile[v][c];
  }
}

// ---------------- K2: conv1x1 via WMMA: h2pre[b,v,o] = sum_i x2[b,i,v]*conv_w[o,i] ----------------
// (conv_b skipped: constant-per-channel bias cancels under training-mode BN)
__global__ void __launch_bounds__(256) k_conv1x1(const float* __restrict__ x2,
                                                 const float* __restrict__ conv_w,
                                                 float* __restrict__ h2pre,
                                                 float* __restrict__ part) {
  extern __shared__ float sm[];
  v2f* Bl = (v2f*)sm;              // [cs(32)][nt(4)][hi(2)][row(16)] = 4096 v2f = 32KB
  float* sstat = sm + 8192;        // [16][64][2] = 8KB
  const int tid = threadIdx.x;

  for (int i = tid; i < 4096; i += 256) {
    int row = i & 15, hi = (i >> 4) & 1, nt = (i >> 5) & 3, cs = i >> 7;
    int c = cs * 4 + 2 * hi, o = nt * 16 + row;
    v2f bb; bb.x = conv_w[o * INCH + c]; bb.y = conv_w[o * INCH + c + 1];
    Bl[i] = bb;
  }
  __syncthreads();

  const int w = tid >> 5, l = tid & 31, row = l & 15, hi = l >> 4;
  float ssum[4] = {0.f, 0.f, 0.f, 0.f}, ssq[4] = {0.f, 0.f, 0.f, 0.f};

  for (int gt = blockIdx.x * 8 + w; gt < NTILES; gt += NWG_K2 * 8) {
    int b = gt / TPB, v0 = (gt % TPB) * 16;
    const float* xb = x2 + (size_t)b * INCH * NVv;
    int vr = v0 + row; if (vr >= NVv) vr = NVv - 1;   // clamp; masked at store
    v8f acc[4];
    #pragma unroll
    for (int nt = 0; nt < 4; nt++)
      #pragma unroll
      for (int r = 0; r < 8; r++) acc[nt][r] = 0.f;
    #pragma unroll
    for (int cs = 0; cs < 32; cs++) {
      int c = cs * 4 + 2 * hi;
      v2f a; a.x = xb[(size_t)c * NVv + vr]; a.y = xb[(size_t)(c + 1) * NVv + vr];
      #pragma unroll
      for (int nt = 0; nt < 4; nt++)
        acc[nt] = wmma_f32_k4(a, Bl[cs * 128 + nt * 32 + hi * 16 + row], acc[nt]);
    }
    #pragma unroll
    for (int nt = 0; nt < 4; nt++) {
      int ch = nt * 16 + row;
      #pragma unroll
      for (int r = 0; r < 8; r++) {
        int v = v0 + r + 8 * hi;
        if (v < NVv) {
          float val = acc[nt][r];
          h2pre[((size_t)b * NVv + v) * 64 + ch] = val;
          ssum[nt] += val; ssq[nt] += val * val;
        }
      }
    }
  }
  int slot = w * 2 + hi;
  #pragma unroll
  for (int nt = 0; nt < 4; nt++) {
    int ch = nt * 16 + row;
    sstat[(slot * 64 + ch) * 2 + 0] = ssum[nt];
    sstat[(slot * 64 + ch) * 2 + 1] = ssq[nt];
  }
  __syncthreads();
  if (tid < 64) {
    float s = 0.f, q = 0.f;
    for (int sl = 0; sl < 16; sl++) {
      s += sstat[(sl * 64 + tid) * 2 + 0];
      q += sstat[(sl * 64 + tid) * 2 + 1];
    }
    part[((size_t)blockIdx.x * 64 + tid) * 2 + 0] = s;
    part[((size_t)blockIdx.x * 64 + tid) * 2 + 1] = q;
  }
}

// ---------------- K3: reduce per-wg partials -> BN scale/shift ----------------
__global__ void k_bnstats(const float* __restrict__ part, int nwg,
                          const float* __restrict__ g, const float* __restrict__ bb,
                          float* __restrict__ bnp) {
  int c = threadIdx.x;
  if (c >= 64) return;
  double s = 0.0, q = 0.0;
  for (int i = 0; i < nwg; i++) {
    s += (double)part[((size_t)i * 64 + c) * 2 + 0];
    q += (double)part[((size_t)i * 64 + c) * 2 + 1];
  }
  double n = (double)B_ * (double)NVv;
  double mean = s / n;
  double var = q / n - mean * mean;
  if (var < 0.0) var = 0.0;
  float sc = g[c] * (float)(1.0 / sqrt(var + (double)EPSf));
  bnp[c * 2 + 0] = sc;
  bnp[c * 2 + 1] = bb[c] - (float)mean * sc;
}

// ---------------- row fetch: 8 channels of one vertex ----------------
// MODE 0: x1_pad row (zero beyond NVIN).  MODE 1: relu(bn1(h1pre)) + relu(bn2(h2pre)).
template <int MODE>
__device__ inline void fetch_row(const float* __restrict__ in0, const float* __restrict__ in1,
                                 int b, int idx, int c0, const float* s1, const float* t1,
                                 const float* s2, const float* t2, float* r) {
  if (MODE == 0) {
    if (idx < NVIN) {
      const float4* p = (const float4*)(in0 + ((size_t)b * NVIN + idx) * 64 + c0);
      float4 u = p[0], v = p[1];
      r[0] = u.x; r[1] = u.y; r[2] = u.z; r[3] = u.w;
      r[4] = v.x; r[5] = v.y; r[6] = v.z; r[7] = v.w;
    } else {
      #pragma unroll
      for (int i = 0; i < 8; i++) r[i] = 0.f;
    }
  } else {
    const float4* p = (const float4*)(in0 + ((size_t)b * NVv + idx) * 64 + c0);
    const float4* q = (const float4*)(in1 + ((size_t)b * NVv + idx) * 64 + c0);
    float4 u = p[0], v = p[1], x = q[0], y = q[1];
    float a_[8] = {u.x, u.y, u.z, u.w, v.x, v.y, v.z, v.w};
    float b_[8] = {x.x, x.y, x.z, x.w, y.x, y.y, y.z, y.w};
    #pragma unroll
    for (int i = 0; i < 8; i++) {
      float h1 = s1[i] * a_[i] + t1[i];
      float h2 = s2[i] * b_[i] + t2[i];
      r[i] = (h1 > 0.f ? h1 : 0.f) + (h2 > 0.f ? h2 : 0.f);
    }
  }
}

// ---------------- K4/K5: mesh_conv (feature gather -> [16v x 256k]x[256k x 64o] WMMA GEMM) ----------------
template <int MODE>
__global__ void __launch_bounds__(256) k_meshconv(
    const float* __restrict__ in0, const float* __restrict__ in1,
    const float* __restrict__ bnp1, const float* __restrict__ bnp2,
    const float* __restrict__ coeffs,
    const int* __restrict__ nbr_idx, const float* __restrict__ nbr_w,
    const int* __restrict__ face_idx, const float* __restrict__ gwe,
    const float* __restrict__ gwn, const int* __restrict__ v2f_idx,
    const float* __restrict__ v2f_w, float* __restrict__ outp, float* __restrict__ part) {
  extern __shared__ float sm[];
  // feat: [sub(2)][tap(4)][v(16)][c stride 66]  (pad 66 kills stride-64 LDS bank conflicts)
  float* feat = sm;                               // 2*4*16*66 = 8448 floats
  v2f* Bl = (v2f*)(sm + 8448);                    // [tap(4)][cs(16)][nt(4)][hi(2)][row(16)] = 8192 v2f = 64KB
  float* sstat = sm + 8448 + 16384;               // [16][16][2] = 512 floats
  const int tid = threadIdx.x;

  // pre-swizzle coefficients into B-fragment order: Bmat_tap[c][o] = coeffs[o,c,tap]
  for (int i = tid; i < 8192; i += 256) {
    int row = i & 15, hi = (i >> 4) & 1, nt = (i >> 5) & 3, cs = (i >> 7) & 15, tap = (i >> 11) & 3;
    int c = cs * 4 + 2 * hi, o = nt * 16 + row;
    v2f bb; bb.x = coeffs[o * 256 + c * 4 + tap]; bb.y = coeffs[o * 256 + (c + 1) * 4 + tap];
    Bl[i] = bb;
  }

  const int c0 = (tid & 7) * 8;
  float s1[8], t1[8], s2[8], t2[8];
  if (MODE) {
    #pragma unroll
    for (int i = 0; i < 8; i++) {
      s1[i] = bnp1[(c0 + i) * 2 + 0]; t1[i] = bnp1[(c0 + i) * 2 + 1];
      s2[i] = bnp2[(c0 + i) * 2 + 0]; t2[i] = bnp2[(c0 + i) * 2 + 1];
    }
  }
  const int v32 = tid >> 3, sub = v32 >> 4, vloc = v32 & 15;
  const int w = tid >> 5, l = tid & 31, row = l & 15, hi = l >> 4;
  const int wsub = w >> 2, wnt = w & 3;
  float ssum = 0.f, ssq = 0.f;

  for (int gt0 = blockIdx.x * 2; gt0 < NTILES; gt0 += NWG_MC * 2) {
    __syncthreads();   // feat reuse + (first iter) Bl visibility
    // ---- gather phase: this thread builds 8 channels of one vertex of sub-tile `sub` ----
    float f0[8], f1_[8], f2[8], f3[8];
    #pragma unroll
    for (int i = 0; i < 8; i++) { f0[i] = f1_[i] = f2[i] = f3[i] = 0.f; }
    int gt = gt0 + sub;
    if (gt < NTILES) {
      int b = gt / TPB, v = (gt % TPB) * 16 + vloc;
      if (v < NVv) {
        fetch_row<MODE>(in0, in1, b, v, c0, s1, t1, s2, t2, f0);      // identity tap
        float r[8];
        for (int j = 0; j < 6; j++) {                                  // Laplacian tap
          int nb2 = nbr_idx[v * 6 + j];
          float wg = nbr_w[v * 6 + j];
          fetch_row<MODE>(in0, in1, b, nb2, c0, s1, t1, s2, t2, r);
          #pragma unroll
          for (int i = 0; i < 8; i++) f1_[i] += wg * r[i];
        }
        for (int j = 0; j < 6; j++) {                                  // EW/NS gradient taps
          int f = v2f_idx[v * 6 + j];
          float vw = v2f_w[v * 6 + j];
          for (int k = 0; k < 3; k++) {
            int fv = face_idx[f * 3 + k];
            float we = gwe[f * 3 + k] * vw, wn = gwn[f * 3 + k] * vw;
            fetch_row<MODE>(in0, in1, b, fv, c0, s1, t1, s2, t2, r);
            #pragma unroll
            for (int i = 0; i < 8; i++) { f2[i] += we * r[i]; f3[i] += wn * r[i]; }
          }
        }
      }
    }
    {
      int base = (sub * 4 * 16 + vloc) * 66 + c0;
      #pragma unroll
      for (int i = 0; i < 8; i++) {
        feat[base + i]            = f0[i];
        feat[base + i + 1 * 1056] = f1_[i];
        feat[base + i + 2 * 1056] = f2[i];
        feat[base + i + 3 * 1056] = f3[i];
      }
    }
    __syncthreads();
    // ---- WMMA phase: wave w -> sub-tile wsub, output n-tile wnt; K = 4 taps x 64 ch ----
    int gtw = gt0 + wsub;
    if (gtw < NTILES) {
      v8f acc;
      #pragma unroll
      for (int i = 0; i < 8; i++) acc[i] = 0.f;
      const float* fb = feat + wsub * 4224;
      #pragma unroll
      for (int tap = 0; tap < 4; tap++) {
        #pragma unroll
        for (int cs = 0; cs < 16; cs++) {
          int c = cs * 4 + 2 * hi;
          v2f a = *(const v2f*)(fb + (tap * 16 + row) * 66 + c);
          acc = wmma_f32_k4(a, Bl[tap * 2048 + cs * 128 + wnt * 32 + hi * 16 + row], acc);
        }
      }
      int b = gtw / TPB, v0 = (gtw % TPB) * 16;
      int ch = wnt * 16 + row;
      #pragma unroll
      for (int r2 = 0; r2 < 8; r2++) {
        int v = v0 + r2 + 8 * hi;
        if (v < NVv) {
          float val = acc[r2];
          outp[((size_t)b * NVv + v) * 64 + ch] = val;
          ssum += val; ssq += val * val;
        }
      }
    }
  }
  sstat[((w * 2 + hi) * 16 + row) * 2 + 0] = ssum;
  sstat[((w * 2 + hi) * 16 + row) * 2 + 1] = ssq;
  __syncthreads();
  if (tid < 64) {
    int wnt2 = tid >> 4, row2 = tid & 15;
    float s = 0.f, q = 0.f;
    #pragma unroll
    for (int ws2 = 0; ws2 < 2; ws2++)
      #pragma unroll
      for (int h2 = 0; h2 < 2; h2++) {
        int slot = (wnt2 + ws2 * 4) * 2 + h2;
        s += sstat[(slot * 16 + row2) * 2 + 0];
        q += sstat[(slot * 16 + row2) * 2 + 1];
      }
    part[((size_t)blockIdx.x * 64 + tid) * 2 + 0] = s;
    part[((size_t)blockIdx.x * 64 + tid) * 2 + 1] = q;
  }
}

// ---------------- K7: final BN+ReLU and transpose [B,V,64] -> [B,64,V] ----------------
__global__ void __launch_bounds__(256) k_out(const float* __restrict__ outpre,
                                             const float* __restrict__ bnp,
                                             float* __restrict__ out) {
  __shared__ float tile[64][65];
  const int nb = (NVv + 63) / 64;
  int b = blockIdx.x / nb;
  int v0 = (blockIdx.x % nb) * 64;
  int t = threadIdx.x;
  for (int i = 0; i < 16; i++) {
    int v = i * 4 + (t >> 6), c = t & 63;
    if (v0 + v < NVv) {
      float x = outpre[((size_t)b * NVv + v0 + v) * 64 + c];
      x = bnp[c * 2] * x + bnp[c * 2 + 1];
      tile[v][c] = x > 0.f ? x : 0.f;
    }
  }
  __syncthreads();
  for (int i = 0; i < 16; i++) {
    int c = i * 4 + (t >> 6), v = t & 63;
    if (v0 + v < NVv) out[((size_t)b * 64 + c) * NVv + v0 + v] = tile[v][c];
  }
}

// ---------------- orchestration ----------------
extern "C" void kernel_launch(void* const* d_in, const int* in_sizes, int n_in,
                              void* d_out, int out_size, void* d_ws, size_t ws_size,
                              hipStream_t stream) {
  const float* x1         = (const float*)d_in[0];
  const float* x2         = (const float*)d_in[1];
  const int*   nbr_idx    = (const int*)d_in[2];
  const float* nbr_w      = (const float*)d_in[3];
  const int*   face_idx   = (const int*)d_in[4];
  const float* grad_w     = (const float*)d_in[5];
  const float* ew         = (const float*)d_in[6];
  const float* ns         = (const float*)d_in[7];
  const int*   v2f_idx    = (const int*)d_in[8];
  const float* v2f_w      = (const float*)d_in[9];
  const float* up_coeffs  = (const float*)d_in[10];
  // d_in[11] up_bias: cancels under training-mode BN
  const float* up_bn_g    = (const float*)d_in[12];
  const float* up_bn_b    = (const float*)d_in[13];
  const float* conv_w     = (const float*)d_in[14];
  // d_in[15] conv_b: cancels
  const float* cross_bn_g = (const float*)d_in[16];
  const float* cross_bn_b = (const float*)d_in[17];
  const float* out_coeffs = (const float*)d_in[18];
  // d_in[19] out_bias: cancels
  const float* out_bn_g   = (const float*)d_in[20];
  const float* out_bn_b   = (const float*)d_in[21];
  float* out = (float*)d_out;
  (void)in_sizes; (void)n_in; (void)out_size; (void)ws_size;

  float* ws = (float*)d_ws;
  size_t off = 0;
  auto alloc = [&](size_t n) { float* p = ws + off; off += (n + 63) & ~(size_t)63; return p; };
  float* x1t    = alloc((size_t)B_ * NVIN * 64);
  float* h1pre  = alloc((size_t)B_ * NVv * 64);
  float* h2pre  = alloc((size_t)B_ * NVv * 64);
  float* outpre = alloc((size_t)B_ * NVv * 64);
  float* gwe    = alloc((size_t)NFf * 3);
  float* gwn    = alloc((size_t)NFf * 3);
  float* part1  = alloc((size_t)NWG_MC * 64 * 2);
  float* part2  = alloc((size_t)NWG_K2 * 64 * 2);
  float* part3  = alloc((size_t)NWG_MC * 64 * 2);
  float* bnp1   = alloc(128);
  float* bnp2   = alloc(128);
  float* bnp3   = alloc(128);

  const size_t K2_SMEM = (size_t)(8192 + 2048) * 4;          // 40KB
  const size_t MC_SMEM = (size_t)(8448 + 16384 + 512) * 4;   // ~99KB (dynamic; WGP has 320KB)

  k_prep_face<<<(NFf * 3 + 255) / 256, 256, 0, stream>>>(grad_w, ew, ns, gwe, gwn);
  k_transpose_x1<<<B_ * ((NVIN + 63) / 64), 256, 0, stream>>>(x1, x1t);
  k_conv1x1<<<NWG_K2, 256, K2_SMEM, stream>>>(x2, conv_w, h2pre, part2);
  k_meshconv<0><<<NWG_MC, 256, MC_SMEM, stream>>>(x1t, nullptr, nullptr, nullptr, up_coeffs,
      nbr_idx, nbr_w, face_idx, gwe, gwn, v2f_idx, v2f_w, h1pre, part1);
  k_bnstats<<<1, 64, 0, stream>>>(part1, NWG_MC, up_bn_g, up_bn_b, bnp1);
  k_bnstats<<<1, 64, 0, stream>>>(part2, NWG_K2, cross_bn_g, cross_bn_b, bnp2);
  k_meshconv<1><<<NWG_MC, 256, MC_SMEM, stream>>>(h1pre, h2pre, bnp1, bnp2, out_coeffs,
      nbr_idx, nbr_w, face_idx, gwe, gwn, v2f_idx, v2f_w, outpre, part3);
  k_bnstats<<<1, 64, 0, stream>>>(part3, NWG_MC, out_bn_g, out_bn_b, bnp3);
  k_out<<<B_ * ((NVv + 63) / 64), 256, 0, stream>>>(outpre, bnp3, out);
}